// GraphConvolutionLayer_78804059947399
// MI455X (gfx1250) — compile-verified
//
#include <hip/hip_runtime.h>

typedef __attribute__((ext_vector_type(2))) float v2f;
typedef __attribute__((ext_vector_type(8))) float v8f;

#define F 128  // feature dim

// ---------------- zero the aggregation buffer (d_out used in place) --------
__global__ __launch_bounds__(256) void zero_kernel(float4* __restrict__ out, int n4) {
    int i = blockIdx.x * blockDim.x + threadIdx.x;
    if (i < n4) out[i] = float4{0.f, 0.f, 0.f, 0.f};
}

// ---------------- edge scatter: out[dst[e]] += h[src[e]] -------------------
// One wave handles 32 edges. Lanes cooperatively load 32 (src,dst) pairs,
// then each edge is broadcast via v_readlane so the per-edge feature row
// (512B) is a coalesced float4-per-lane load + 4 f32 L2 atomics per lane.
__global__ __launch_bounds__(256) void scatter_kernel(const float* __restrict__ h,
                                                      const int* __restrict__ src,
                                                      const int* __restrict__ dst,
                                                      float* __restrict__ out, int E) {
    int tid  = blockIdx.x * blockDim.x + threadIdx.x;
    int lane = tid & 31;
    int base = (tid >> 5) << 5;            // this wave's first edge
    if (base >= E) return;
    int nvalid = E - base;
    if (nvalid > 32) nvalid = 32;

    int s = 0, d = 0;
    if (base + lane < E) { s = src[base + lane]; d = dst[base + lane]; }

    for (int e = 0; e < nvalid; ++e) {
        int se = __builtin_amdgcn_readlane(s, e);   // uniform (SGPR) node ids
        int de = __builtin_amdgcn_readlane(d, e);
        float4 v = reinterpret_cast<const float4*>(h + (size_t)se * F)[lane];
        float* op = out + (size_t)de * F + lane * 4;
        atomicAdd(op + 0, v.x);
        atomicAdd(op + 1, v.y);
        atomicAdd(op + 2, v.z);
        atomicAdd(op + 3, v.w);
    }
}

// ---------------- in-place GEMM + bias + ReLU via V_WMMA_F32_16X16X4_F32 ---
// out[i,:] = relu(agg[i,:] @ W^T + b).  Each wave owns a 16-row stripe:
// A-tile (16x128 f32) cached in 64 VGPRs, W staged in 64KB LDS with an XOR
// swizzle for conflict-free ds_load_b64 B-fragment reads.
__global__ __launch_bounds__(256) void gemm_relu_kernel(float* __restrict__ out,
                                                        const float* __restrict__ W,
                                                        const float* __restrict__ b,
                                                        int n_nodes) {
    __shared__ float Wl[F * F];  // 64KB, swizzled: Wl[j*128 + (k ^ ((j&15)<<2))] = W[j][k]

    for (int idx = threadIdx.x; idx < (F * F) / 4; idx += 256) {
        int j = idx >> 5;                 // row of W
        int q = idx & 31;                 // float4 within row
        float4 w4 = reinterpret_cast<const float4*>(W)[idx];
        int pos = j * F + ((q * 4) ^ ((j & 15) << 2));   // stays 16B aligned
        *reinterpret_cast<float4*>(&Wl[pos]) = w4;
    }
    __syncthreads();

    int wave    = threadIdx.x >> 5;
    int lane    = threadIdx.x & 31;
    int stripe  = blockIdx.x * 8 + wave;
    int rowbase = stripe * 16;
    if (rowbase >= n_nodes) return;

    // A fragment addressing (16x4 f32 layout): lanes 0-15 -> K={k0,k0+1},
    // lanes 16-31 -> K={k0+2,k0+3}; M = lane&15.
    int m    = lane & 15;
    int koff = (lane >> 4) * 2;
    int arow = rowbase + m;
    if (arow > n_nodes - 1) arow = n_nodes - 1;   // clamp: keep EXEC all-1s

    v2f afrag[32];
#pragma unroll
    for (int kk = 0; kk < 32; ++kk)
        afrag[kk] = *reinterpret_cast<const v2f*>(out + (size_t)arow * F + kk * 4 + koff);

    int swz = m << 2;
    for (int nt = 0; nt < 8; ++nt) {
        int nbase = nt * 16;
        v8f acc = {};
#pragma unroll
        for (int kk = 0; kk < 32; ++kk) {
            // B[k][n] = W[nbase+n][k]; swizzled LDS read, 8B aligned, conflict-free
            const float* bp = &Wl[(nbase + m) * F + ((kk * 4 + koff) ^ swz)];
            v2f bfrag = *reinterpret_cast<const v2f*>(bp);
            acc = __builtin_amdgcn_wmma_f32_16x16x4_f32(
                false, afrag[kk], false, bfrag, (short)0, acc, false, false);
        }
        // C/D layout: VGPR v -> M = v (+8 for lanes 16-31), N = lane&15
        float bias = b[nbase + m];
#pragma unroll
        for (int v = 0; v < 8; ++v) {
            int row = rowbase + v + ((lane >> 4) << 3);
            if (row < n_nodes) {
                float r = acc[v] + bias;
                out[(size_t)row * F + nbase + m] = r > 0.f ? r : 0.f;
            }
        }
    }
}

extern "C" void kernel_launch(void* const* d_in, const int* in_sizes, int n_in,
                              void* d_out, int out_size, void* d_ws, size_t ws_size,
                              hipStream_t stream) {
    const float* h  = (const float*)d_in[0];
    const int*  src = (const int*)d_in[1];
    const int*  dst = (const int*)d_in[2];
    const float* W  = (const float*)d_in[3];
    const float* b  = (const float*)d_in[4];
    float* out = (float*)d_out;

    int n_nodes = in_sizes[0] / F;
    int E       = in_sizes[1];

    int n4 = out_size / 4;
    zero_kernel<<<(n4 + 255) / 256, 256, 0, stream>>>((float4*)out, n4);

    scatter_kernel<<<(E + 255) / 256, 256, 0, stream>>>(h, src, dst, out, E);

    int stripes = (n_nodes + 15) / 16;
    int blocks  = (stripes + 7) / 8;     // 8 waves/block, 16 rows/wave
    gemm_relu_kernel<<<blocks, 256, 0, stream>>>(out, W, b, n_nodes);
}